// TSPModel_5119601017349
// MI455X (gfx1250) — compile-verified
//
#include <hip/hip_runtime.h>
#include <hip/hip_bf16.h>
#include <stdint.h>

typedef __attribute__((ext_vector_type(2))) float v2f;
typedef __attribute__((ext_vector_type(8))) float v8f;

#define N_NODES 1024
#define HIDDEN  128

// ---------------------------------------------------------------------------
// K1: per-row positive/negative sums of edge_weights (the only 4MB read).
// One wave32 per row; float4 vector loads; shfl_xor cross-lane reduce.
// ---------------------------------------------------------------------------
__global__ __launch_bounds__(256)
void k_rowsum(const float* __restrict__ e, float* __restrict__ P,
              float* __restrict__ M) {
  const int wave = threadIdx.x >> 5;
  const int lane = threadIdx.x & 31;
  const int row  = blockIdx.x * 8 + wave;
  const float4* erow = (const float4*)(e + (size_t)row * N_NODES);
  float p = 0.f, m = 0.f;
#pragma unroll
  for (int i = 0; i < 8; ++i) {
    float4 v = erow[lane + i * 32];
    p += fmaxf(v.x, 0.f) + fmaxf(v.y, 0.f) + fmaxf(v.z, 0.f) + fmaxf(v.w, 0.f);
    m += fminf(v.x, 0.f) + fminf(v.y, 0.f) + fminf(v.z, 0.f) + fminf(v.w, 0.f);
  }
#pragma unroll
  for (int s = 16; s >= 1; s >>= 1) {
    p += __shfl_xor(p, s, 32);
    m += __shfl_xor(m, s, 32);
  }
  if (lane == 0) { P[row] = p; M[row] = m; }
}

// ---------------------------------------------------------------------------
// K2: single-block fused "small" math.
//   c1[j] = sum_k max(t4,0)*t3[j,k] ; c2[j] = sum_k min(t4,0)*t3[j,k]
//   musum[j]  = sum_u relu(na*t1 + P*c1 + M*c2)         (mu1 colsum, no mu1)
//   nb[i]     = theta2[i,:] @ musum
//   mu2sum[j] = sum_u relu(na*t1 + nb + P*c1 + M*c2)
//   C         = sum_i relu(theta6[i,:] @ mu2sum) * t5[128+i]
// 1024 threads: tid = g*128 + j, g in [0,8) splits the u-loop.
// ---------------------------------------------------------------------------
__global__ __launch_bounds__(1024)
void k_small(const uint8_t* __restrict__ na, const float* __restrict__ t1,
             const float* __restrict__ t2, const float* __restrict__ t3,
             const float* __restrict__ t4, const float* __restrict__ t5,
             const float* __restrict__ t6, const float* __restrict__ P,
             const float* __restrict__ M, float* __restrict__ c1o,
             float* __restrict__ c2o, float* __restrict__ nbo,
             float* __restrict__ Co) {
  __shared__ float sc1[HIDDEN], sc2[HIDDEN], snb[HIDDEN];
  __shared__ float part[8][HIDDEN];
  __shared__ float red[HIDDEN];
  const int tid = threadIdx.x;
  const int j = tid & (HIDDEN - 1);
  const int g = tid >> 7;

  if (tid < HIDDEN) {
    float a = 0.f, b = 0.f;
    for (int k = 0; k < HIDDEN; ++k) {
      float t = t4[k], w = t3[tid * HIDDEN + k];
      a = fmaf(fmaxf(t, 0.f), w, a);
      b = fmaf(fminf(t, 0.f), w, b);
    }
    sc1[tid] = a; sc2[tid] = b;
  }
  __syncthreads();

  const float c1j = sc1[j], c2j = sc2[j], t1j = t1[j];

  // mu1 column sum
  float acc = 0.f;
  for (int u = g * 128; u < (g + 1) * 128; ++u) {
    float an = (float)na[u];
    float t = fmaf(P[u], c1j, fmaf(M[u], c2j, an * t1j));
    acc += fmaxf(t, 0.f);
  }
  part[g][j] = acc;
  __syncthreads();
  if (tid < HIDDEN) {
    float s = 0.f;
#pragma unroll
    for (int gg = 0; gg < 8; ++gg) s += part[gg][tid];
    red[tid] = s;                       // musum
  }
  __syncthreads();
  if (tid < HIDDEN) {
    float s = 0.f;
    for (int k = 0; k < HIDDEN; ++k) s = fmaf(t2[tid * HIDDEN + k], red[k], s);
    snb[tid] = s;                       // neighbor embedding (iter 2)
  }
  __syncthreads();

  const float nbj = snb[j];
  // mu2 column sum
  acc = 0.f;
  for (int u = g * 128; u < (g + 1) * 128; ++u) {
    float an = (float)na[u];
    float t = fmaf(P[u], c1j, fmaf(M[u], c2j, fmaf(an, t1j, nbj)));
    acc += fmaxf(t, 0.f);
  }
  part[g][j] = acc;
  __syncthreads();
  if (tid < HIDDEN) {
    float s = 0.f;
#pragma unroll
    for (int gg = 0; gg < 8; ++gg) s += part[gg][tid];
    red[tid] = s;                       // mu2sum
  }
  __syncthreads();
  if (tid < HIDDEN) {
    float s = 0.f;
    for (int k = 0; k < HIDDEN; ++k) s = fmaf(t6[tid * HIDDEN + k], red[k], s);
    part[0][tid] = fmaxf(s, 0.f) * t5[HIDDEN + tid];  // graph-embedding term
  }
  __syncthreads();
  if (tid < 64) part[0][tid] += part[0][tid + 64];
  __syncthreads();
  if (tid < 32) {
    float v = part[0][tid] + part[0][tid + 32];
#pragma unroll
    for (int s = 16; s >= 1; s >>= 1) v += __shfl_xor(v, s, 32);
    if (tid == 0) Co[0] = v;
  }
  if (tid < HIDDEN) { c1o[tid] = sc1[tid]; c2o[tid] = sc2[tid]; nbo[tid] = snb[tid]; }
}

// ---------------------------------------------------------------------------
// K3: head GEMM via fp32 WMMA:  ne = mu2 @ theta7^T, fused epilogue
//     q[u] = sum_k relu(ne[u,k]) * t5[k] + C.
// mu2 A-fragments built in-register from (na,P,M) + LDS-staged per-j params.
// A layout (16x4 f32): lanes 0-15 hold K=0,1; lanes 16-31 hold K=2,3.
// Each wave: 16 u-rows, 8 N-tiles held live, 32 K-steps -> 256 v_wmma ops.
// ---------------------------------------------------------------------------
__global__ __launch_bounds__(128)
void k_head(const uint8_t* __restrict__ na, const float* __restrict__ t1,
            const float* __restrict__ t5, const float* __restrict__ t7,
            const float* __restrict__ P, const float* __restrict__ M,
            const float* __restrict__ c1, const float* __restrict__ c2,
            const float* __restrict__ nb, const float* __restrict__ Co,
            float* __restrict__ q) {
  __shared__ float4 prm[HIDDEN];     // {theta1, nb, c1, c2} per j
  const int tid = threadIdx.x;
  if (tid < HIDDEN)
    prm[tid] = make_float4(t1[tid], nb[tid], c1[tid], c2[tid]);
  __syncthreads();

  const int wave = tid >> 5;
  const int lane = tid & 31;
  const int lo = lane & 15, hi = lane >> 4;
  const int u0 = (blockIdx.x * 4 + wave) * 16;
  const int u = u0 + lo;
  const float an = (float)na[u];
  const float Pu = P[u], Mu = M[u];

  v8f acc[8] = {};
  for (int kk = 0; kk < 32; ++kk) {
    const int j0 = kk * 4 + hi * 2;
    float4 pa = prm[j0];
    float4 pb = prm[j0 + 1];
    v2f a;
    a.x = fmaxf(fmaf(Pu, pa.z, fmaf(Mu, pa.w, fmaf(an, pa.x, pa.y))), 0.f);
    a.y = fmaxf(fmaf(Pu, pb.z, fmaf(Mu, pb.w, fmaf(an, pb.x, pb.y))), 0.f);
#pragma unroll
    for (int nt = 0; nt < 8; ++nt) {
      const float* bp = t7 + (size_t)(nt * 16 + lo) * HIDDEN + j0;
      v2f b; b.x = bp[0]; b.y = bp[1];
      acc[nt] = __builtin_amdgcn_wmma_f32_16x16x4_f32(
          false, a, false, b, (short)0, acc[nt], false, false);
    }
  }

  const float C = Co[0];
  float p[8];
#pragma unroll
  for (int r = 0; r < 8; ++r) p[r] = 0.f;
#pragma unroll
  for (int nt = 0; nt < 8; ++nt) {
    float w5 = t5[nt * 16 + lo];
#pragma unroll
    for (int r = 0; r < 8; ++r)
      p[r] = fmaf(fmaxf(acc[nt][r], 0.f), w5, p[r]);
  }
#pragma unroll
  for (int r = 0; r < 8; ++r) {
#pragma unroll
    for (int s = 8; s >= 1; s >>= 1) p[r] += __shfl_xor(p[r], s, 32);
  }
  if (lo == 0) {
#pragma unroll
    for (int r = 0; r < 8; ++r)
      q[u0 + hi * 8 + r] = p[r] + C;
  }
}

// ---------------------------------------------------------------------------
extern "C" void kernel_launch(void* const* d_in, const int* in_sizes, int n_in,
                              void* d_out, int out_size, void* d_ws, size_t ws_size,
                              hipStream_t stream) {
  const uint8_t* na = (const uint8_t*)d_in[0];   // jnp.bool_ -> 1 byte/elem
  const float* e  = (const float*)d_in[1];
  const float* t1 = (const float*)d_in[2];
  const float* t2 = (const float*)d_in[3];
  const float* t3 = (const float*)d_in[4];
  const float* t4 = (const float*)d_in[5];
  const float* t5 = (const float*)d_in[6];
  const float* t6 = (const float*)d_in[7];
  const float* t7 = (const float*)d_in[8];
  float* ws = (float*)d_ws;
  float* P  = ws;            // 1024
  float* M  = ws + 1024;     // 1024
  float* c1 = ws + 2048;     // 128
  float* c2 = ws + 2176;     // 128
  float* nb = ws + 2304;     // 128
  float* Co = ws + 2432;     // 1
  float* q  = (float*)d_out; // 1024

  k_rowsum<<<128, 256, 0, stream>>>(e, P, M);
  k_small<<<1, 1024, 0, stream>>>(na, t1, t2, t3, t4, t5, t6, P, M, c1, c2, nb, Co);
  k_head<<<16, 128, 0, stream>>>(na, t1, t5, t7, P, M, c1, c2, nb, Co, q);
}